// Involution2D_40415642255609
// MI455X (gfx1250) — compile-verified
//
#include <hip/hip_runtime.h>

// CDNA5 wave32 WMMA operand vectors for V_WMMA_F32_16X16X4_F32:
// A (16x4 f32) and B (4x16 f32) = 2 VGPRs/lane; C/D (16x16 f32) = 8 VGPRs/lane.
typedef __attribute__((ext_vector_type(2))) float v2f;
typedef __attribute__((ext_vector_type(8))) float v8f;

#define KS     7
#define GC     16
#define GROUPS 16
#define CIN    256
#define MID    64
#define K2     49      // KS*KS
#define WOUT   784     // K2*GROUPS
#define NB     4
#define NH     56
#define NW     56
#define HW     3136    // NH*NW
#define NPIX   12544   // NB*HW  (== 784 tiles of 16)
#define NT     784     // NPIX/16
#define TR     4       // row-tile for the apply kernel

// ---------------------------------------------------------------------------
// Stage 1: t[64, N] = w1[64x256] * X[256, N] + b1   (WMMA f32, K=256)
// One wave handles one 16-wide N-tile and all 4 M-tiles (B operand shared 4x).
// ---------------------------------------------------------------------------
__global__ __launch_bounds__(256) void invol_gemm1(
    const float* __restrict__ x, const float* __restrict__ w1,
    const float* __restrict__ b1, float* __restrict__ t)
{
  int wave = (blockIdx.x * blockDim.x + threadIdx.x) >> 5;   // wave32
  if (wave >= NT) return;                                    // wave-uniform
  int lane = threadIdx.x & 31;
  int nn   = lane & 15;          // N within tile
  int hv   = lane >> 4;          // lane half
  int kb   = hv << 1;            // K sub-offset {0,2}
  int m    = lane & 15;          // M within tile (A operand)

  int n    = wave * 16 + nn;
  int bimg = n / HW;
  int rem  = n - bimg * HW;      // whole 16-tile stays inside one image (HW%16==0)
  const float* xb = x + (size_t)bimg * CIN * HW + rem;       // + c*HW per channel

  v8f c0 = {}, c1 = {}, c2 = {}, c3 = {};
  for (int k0 = 0; k0 < CIN; k0 += 4) {
    v2f bm;
    bm.x = xb[(size_t)(k0 + kb) * HW];
    bm.y = xb[(size_t)(k0 + kb + 1) * HW];
    const float* wr = w1 + k0 + kb;
    v2f a0, a1, a2, a3;
    a0.x = wr[(m +  0) * CIN]; a0.y = wr[(m +  0) * CIN + 1];
    a1.x = wr[(m + 16) * CIN]; a1.y = wr[(m + 16) * CIN + 1];
    a2.x = wr[(m + 32) * CIN]; a2.y = wr[(m + 32) * CIN + 1];
    a3.x = wr[(m + 48) * CIN]; a3.y = wr[(m + 48) * CIN + 1];
    c0 = __builtin_amdgcn_wmma_f32_16x16x4_f32(false, a0, false, bm, (short)0, c0, false, false);
    c1 = __builtin_amdgcn_wmma_f32_16x16x4_f32(false, a1, false, bm, (short)0, c1, false, false);
    c2 = __builtin_amdgcn_wmma_f32_16x16x4_f32(false, a2, false, bm, (short)0, c2, false, false);
    c3 = __builtin_amdgcn_wmma_f32_16x16x4_f32(false, a3, false, bm, (short)0, c3, false, false);
  }
  int mh = hv * 8;   // C/D row base for this lane half
  #pragma unroll
  for (int r = 0; r < 8; ++r) {
    t[(size_t)( 0 + mh + r) * NPIX + n] = c0[r] + b1[ 0 + mh + r];
    t[(size_t)(16 + mh + r) * NPIX + n] = c1[r] + b1[16 + mh + r];
    t[(size_t)(32 + mh + r) * NPIX + n] = c2[r] + b1[32 + mh + r];
    t[(size_t)(48 + mh + r) * NPIX + n] = c3[r] + b1[48 + mh + r];
  }
}

// ---------------------------------------------------------------------------
// Stage 2a: per-channel batch stats over N; emit fused scale/shift:
//   stats[c]       = gamma[c] * rsqrt(var + eps)
//   stats[MID + c] = beta[c] - mean * scale
// ---------------------------------------------------------------------------
__global__ __launch_bounds__(256) void invol_bnstats(
    const float* __restrict__ t, const float* __restrict__ gamma,
    const float* __restrict__ beta, float* __restrict__ stats)
{
  int ch = blockIdx.x;
  const float* row = t + (size_t)ch * NPIX;
  float s = 0.f, ss = 0.f;
  for (int i = threadIdx.x; i < NPIX; i += 256) {
    float v = row[i];
    s += v; ss += v * v;
  }
  __shared__ float rs[256], rss[256];
  rs[threadIdx.x] = s; rss[threadIdx.x] = ss;
  __syncthreads();
  for (int off = 128; off > 0; off >>= 1) {
    if (threadIdx.x < off) {
      rs[threadIdx.x]  += rs[threadIdx.x + off];
      rss[threadIdx.x] += rss[threadIdx.x + off];
    }
    __syncthreads();
  }
  if (threadIdx.x == 0) {
    float mean = rs[0] * (1.0f / NPIX);
    float var  = rss[0] * (1.0f / NPIX) - mean * mean;
    float sc   = gamma[ch] * rsqrtf(var + 1e-5f);
    stats[ch]       = sc;
    stats[MID + ch] = beta[ch] - mean * sc;
  }
}

// ---------------------------------------------------------------------------
// Stage 2b: apply BN affine + ReLU ONCE into tn[64, N]
// (avoids recomputing it 49x inside the stage-3 GEMM).
// ---------------------------------------------------------------------------
__global__ __launch_bounds__(256) void invol_bnapply(
    const float* __restrict__ t, const float* __restrict__ stats,
    float* __restrict__ tn)
{
  int ch   = blockIdx.x >> 3;                 // 64 channels x 8 slices
  int part = blockIdx.x & 7;
  float sc = stats[ch], sh = stats[MID + ch];
  const float* src = t  + (size_t)ch * NPIX + part * (NPIX / 8);
  float*       dst = tn + (size_t)ch * NPIX + part * (NPIX / 8);
  for (int i = threadIdx.x; i < NPIX / 8; i += 256)
    dst[i] = fmaxf(src[i] * sc + sh, 0.f);
}

// ---------------------------------------------------------------------------
// Stage 3: wgt[784, N] = w2[784x64] * tn[64, N] + b2  (WMMA f32, K=64)
// 49 M-tiles = 7x7: each wave owns 7 M-tiles for one N-tile, so the B operand
// (tn column) is loaded once per K-step and reused across 7 independent
// back-to-back WMMAs (no accumulator RAW hazards -> full XDL pipelining).
// ---------------------------------------------------------------------------
__global__ __launch_bounds__(256) void invol_gemm2(
    const float* __restrict__ tn, const float* __restrict__ w2,
    const float* __restrict__ b2, float* __restrict__ wgt)
{
  int wave = (blockIdx.x * blockDim.x + threadIdx.x) >> 5;
  if (wave >= 7 * NT) return;                                // wave-uniform
  int mg = wave / NT;            // which group of 7 M-tiles (0..6)
  int nt = wave - mg * NT;
  int lane = threadIdx.x & 31;
  int nn = lane & 15;
  int hv = lane >> 4;
  int kb = hv << 1;
  int m  = lane & 15;
  int n     = nt * 16 + nn;
  int mrow0 = mg * 7 * 16;       // first output row of this wave's M-tiles

  v8f acc[7] = {};
  const float* wbase = w2 + (size_t)(mrow0 + m) * MID + kb;
  for (int k0 = 0; k0 < MID; k0 += 4) {
    v2f bm;
    bm.x = tn[(size_t)(k0 + kb) * NPIX + n];
    bm.y = tn[(size_t)(k0 + kb + 1) * NPIX + n];
    #pragma unroll
    for (int j = 0; j < 7; ++j) {
      const float* wp = wbase + (size_t)j * 16 * MID + k0;
      v2f am;
      am.x = wp[0];
      am.y = wp[1];
      acc[j] = __builtin_amdgcn_wmma_f32_16x16x4_f32(false, am, false, bm, (short)0, acc[j], false, false);
    }
  }
  int mh = hv * 8;
  #pragma unroll
  for (int j = 0; j < 7; ++j) {
    int rowb = mrow0 + j * 16 + mh;
    #pragma unroll
    for (int r = 0; r < 8; ++r)
      wgt[(size_t)(rowb + r) * NPIX + n] = acc[j][r] + b2[rowb + r];
  }
}

// ---------------------------------------------------------------------------
// Stage 4: involution apply. Block = (b, g, 4-row strip). The 16-channel
// padded x patch lives in LDS laid out [row][col][channel] with stride 17
// floats so the 32 lanes (which differ in col) hit 32 distinct banks.
// wgt values have zero cross-thread reuse -> streamed straight from L2.
// ---------------------------------------------------------------------------
__global__ __launch_bounds__(256) void invol_apply(
    const float* __restrict__ x, const float* __restrict__ wgt,
    float* __restrict__ out)
{
  int bid = blockIdx.x;
  int rt  = bid % (NH / TR);
  int g   = (bid / (NH / TR)) % GROUPS;
  int b   = bid / ((NH / TR) * GROUPS);
  int r0  = rt * TR;

  __shared__ float xs[(TR + 6) * (NW + 6) * 17];   // 10*62*17 floats = 42.2 KB
  const float* xg = x + ((size_t)b * CIN + g * GC) * HW;

  for (int idx = threadIdx.x; idx < GC * (TR + 6) * (NW + 6); idx += 256) {
    int c  = idx / ((TR + 6) * (NW + 6));
    int rc = idx - c * ((TR + 6) * (NW + 6));
    int rr = rc / (NW + 6);
    int cc = rc - rr * (NW + 6);
    int gr = r0 + rr - 3;
    int gl = cc - 3;
    float v = 0.f;
    if (gr >= 0 && gr < NH && gl >= 0 && gl < NW)
      v = xg[(size_t)c * HW + gr * NW + gl];
    xs[(rr * (NW + 6) + cc) * 17 + c] = v;
  }
  __syncthreads();

  int pid = threadIdx.x;
  if (pid < TR * NW) {
    int r   = pid / NW;
    int col = pid - r * NW;
    int n   = b * HW + (r0 + r) * NW + col;
    float acc[GC];
    #pragma unroll
    for (int c = 0; c < GC; ++c) acc[c] = 0.f;

    const float* wg = wgt + (size_t)(g * K2) * NPIX + n;
    for (int k = 0; k < K2; ++k) {
      int di = k / KS, dj = k - di * KS;
      float w = wg[(size_t)k * NPIX];
      const float* xp = &xs[((r + di) * (NW + 6) + (col + dj)) * 17];
      #pragma unroll
      for (int c = 0; c < GC; ++c) acc[c] += w * xp[c];
    }
    float* og = out + ((size_t)b * CIN + g * GC) * HW + (r0 + r) * NW + col;
    #pragma unroll
    for (int c = 0; c < GC; ++c) og[(size_t)c * HW] = acc[c];
  }
}

// ---------------------------------------------------------------------------
extern "C" void kernel_launch(void* const* d_in, const int* in_sizes, int n_in,
                              void* d_out, int out_size, void* d_ws, size_t ws_size,
                              hipStream_t stream)
{
  const float* x     = (const float*)d_in[0];
  const float* w1    = (const float*)d_in[1];
  const float* b1    = (const float*)d_in[2];
  const float* gamma = (const float*)d_in[3];
  const float* beta  = (const float*)d_in[4];
  const float* w2    = (const float*)d_in[5];
  const float* b2    = (const float*)d_in[6];
  float* out = (float*)d_out;

  float* t     = (float*)d_ws;                    // 64 * 12544 floats (3.2 MB)
  float* stats = t + (size_t)MID * NPIX;          // 128 floats
  float* tn    = stats + 2 * MID;                 // 64 * 12544 floats (3.2 MB)
  float* wgt   = tn + (size_t)MID * NPIX;         // 784 * 12544 floats (39.3 MB)

  invol_gemm1  <<<NT / 8,                 256, 0, stream>>>(x, w1, b1, t);
  invol_bnstats<<<MID,                    256, 0, stream>>>(t, gamma, beta, stats);
  invol_bnapply<<<MID * 8,                256, 0, stream>>>(t, stats, tn);
  invol_gemm2  <<<(7 * NT) / 8,           256, 0, stream>>>(tn, w2, b2, wgt);
  invol_apply  <<<NB * GROUPS * (NH / TR), 256, 0, stream>>>(x, wgt, out);
}